// BoundingBoxExtractor_41729902248169
// MI455X (gfx1250) — compile-verified
//
#include <hip/hip_runtime.h>
#include <stdint.h>

// Problem geometry (fixed by the reference: mask [32, 8, 512, 512] f32)
#define HH 512
#define WW 512
#define ROWS 32                 // rows per TDM chunk: 32*512*4B = 64KB per buffer
#define NCHUNK (HH / ROWS)      // 16 chunks per channel
#define NTHREADS 256            // 8 wave32 waves

typedef unsigned int v4u __attribute__((ext_vector_type(4)));
typedef int          v8i __attribute__((ext_vector_type(8)));
typedef int          v4i __attribute__((ext_vector_type(4)));

__device__ __forceinline__ void tdm_issue_chunk(const float* src, int chunk,
                                                unsigned lds0, unsigned lds1,
                                                v8i g1, v4i gz4, v8i gz8) {
    unsigned long long ga =
        (unsigned long long)(uintptr_t)(src + (size_t)chunk * ROWS * WW);
    v4u g0;
    g0[0] = 1u;                                   // count=1, user mode
    g0[1] = (chunk & 1) ? lds1 : lds0;            // lds_addr (bytes)
    g0[2] = (unsigned)(ga & 0xFFFFFFFFull);       // global_addr[31:0]
    g0[3] = (unsigned)((ga >> 32) & 0x1FFFFFFull) // global_addr[56:32]
            | (2u << 30);                         // type = 2 ("image")
    // amdgpu-toolchain (clang-23) 6-arg TDM builtin; cpol = 0
    __builtin_amdgcn_tensor_load_to_lds(g0, g1, gz4, gz4, gz8, 0);
}

__global__ __launch_bounds__(NTHREADS)
void bbox_extract_kernel(const float* __restrict__ mask,
                         int* __restrict__ obj,          // [B*C] int32
                         float* __restrict__ boxes,      // [B*C*4] scaled, then [B*C*4] *2
                         int channels) {
    __shared__ __align__(16) float buf[2][ROWS * WW];  // 128 KB double buffer (of 320 KB/WGP)
    __shared__ float red[NTHREADS / 32][4];

    const int ch = blockIdx.x;            // b*C + c
    if (ch >= channels) return;
    const float* src = mask + (size_t)ch * HH * WW;

    // Flat shared pointers: aperture base has zero low 32 bits -> low32 == LDS byte offset
    const unsigned lds0 = (unsigned)(uintptr_t)&buf[0][0];
    const unsigned lds1 = (unsigned)(uintptr_t)&buf[1][0];

    // D# group 1: data_size=4B (bits17:16=2), tensor_dim0=512 (bits79:48),
    // tensor_dim1=512 (bits111:80), tile_dim0=512 (bits127:112),
    // tile_dim1=ROWS (bits143:128), tensor_dim0_stride=512 (bits207:160)
    v8i g1;
    g1[0] = 0x20000;
    g1[1] = (int)((unsigned)WW << 16);
    g1[2] = (int)((unsigned)HH << 16);
    g1[3] = (int)((unsigned)WW << 16);
    g1[4] = ROWS;
    g1[5] = WW;
    g1[6] = 0;
    g1[7] = 0;
    v4i gz4 = {0, 0, 0, 0};
    v8i gz8 = {0, 0, 0, 0, 0, 0, 0, 0};

    // Scalar wave id so the TDM issue is a scalar branch (TDM ignores EXEC).
    const int wave = __builtin_amdgcn_readfirstlane((int)(threadIdx.x >> 5));

    if (wave == 0) tdm_issue_chunk(src, 0, lds0, lds1, g1, gz4, gz8);

    float xmin = 3.4028235e38f, ymin = 3.4028235e38f;
    float xmax = 0.0f,          ymax = 0.0f;

    for (int k = 0; k < NCHUNK; ++k) {
        __syncthreads();   // buffer (k+1)&1 free for overwrite
        if (wave == 0) {
            if (k + 1 < NCHUNK) {
                tdm_issue_chunk(src, k + 1, lds0, lds1, g1, gz4, gz8);
                __builtin_amdgcn_s_wait_tensorcnt(1);   // chunk k landed (in-order)
            } else {
                __builtin_amdgcn_s_wait_tensorcnt(0);   // last chunk landed
            }
        }
        __syncthreads();   // buffer k&1 visible to all waves

        const float4* b4 = (const float4*)buf[k & 1];
        const float rowbase = (float)(k * ROWS);
        #pragma unroll 4
        for (int i = (int)threadIdx.x; i < (ROWS * WW) / 4; i += NTHREADS) {
            float4 v = b4[i];                 // ds_load_b128; quad never crosses a row (512 % 4 == 0)
            int   base = i << 2;
            int   c = base & (WW - 1);
            int   r = base >> 9;              // WW == 512
            float yrow = rowbase + (float)(r + 1);
            float fc   = (float)c;
            bool s0 = (v.x >= 0.5f), s1 = (v.y >= 0.5f);
            bool s2 = (v.z >= 0.5f), s3 = (v.w >= 0.5f);
            bool all4 = s0 && s1 && s2 && s3;
            bool any4 = s0 || s1 || s2 || s3;
            // min over quad: 0 unless all four set (then smallest col / the row)
            xmin = fminf(xmin, all4 ? fc + 1.0f : 0.0f);
            ymin = fminf(ymin, all4 ? yrow : 0.0f);
            // max over quad: row if any set; highest set column
            ymax = fmaxf(ymax, any4 ? yrow : 0.0f);
            float xm = s3 ? fc + 4.0f
                     : (s2 ? fc + 3.0f
                     : (s1 ? fc + 2.0f
                     : (s0 ? fc + 1.0f : 0.0f)));
            xmax = fmaxf(xmax, xm);
        }
    }

    // wave32 butterfly reduction
    #pragma unroll
    for (int off = 16; off > 0; off >>= 1) {
        xmin = fminf(xmin, __shfl_xor(xmin, off, 32));
        ymin = fminf(ymin, __shfl_xor(ymin, off, 32));
        xmax = fmaxf(xmax, __shfl_xor(xmax, off, 32));
        ymax = fmaxf(ymax, __shfl_xor(ymax, off, 32));
    }
    if ((threadIdx.x & 31) == 0) {
        int w = (int)(threadIdx.x >> 5);
        red[w][0] = xmin; red[w][1] = ymin; red[w][2] = xmax; red[w][3] = ymax;
    }
    __syncthreads();

    if (threadIdx.x == 0) {
        float xm = red[0][0], ym = red[0][1], xM = red[0][2], yM = red[0][3];
        #pragma unroll
        for (int w = 1; w < NTHREADS / 32; ++w) {
            xm = fminf(xm, red[w][0]); ym = fminf(ym, red[w][1]);
            xM = fmaxf(xM, red[w][2]); yM = fmaxf(yM, red[w][3]);
        }
        obj[ch] = ((yM > ym) && (xM > xm)) ? 1 : 0;
        float* bs = boxes + (size_t)ch * 4;              // bounding_boxes_scaled
        bs[0] = xm; bs[1] = ym; bs[2] = xM; bs[3] = yM;
        float* bb = boxes + (size_t)channels * 4 + (size_t)ch * 4;  // * SCALE(2)
        bb[0] = 2.0f * xm; bb[1] = 2.0f * ym; bb[2] = 2.0f * xM; bb[3] = 2.0f * yM;
    }
}

extern "C" void kernel_launch(void* const* d_in, const int* in_sizes, int n_in,
                              void* d_out, int out_size, void* d_ws, size_t ws_size,
                              hipStream_t stream) {
    (void)n_in; (void)out_size; (void)d_ws; (void)ws_size;
    const float* mask = (const float*)d_in[0];
    const int channels = in_sizes[0] / (HH * WW);   // 32*8 = 256

    // d_out layout (tuple flattened in return order):
    //   [0, channels)                      object_found  (int32)
    //   [channels, channels + 4*channels)  bounding_boxes_scaled (f32)
    //   [.., + 4*channels)                 bounding_boxes (f32)
    int*   obj   = (int*)d_out;
    float* boxes = (float*)d_out + channels;

    bbox_extract_kernel<<<channels, NTHREADS, 0, stream>>>(mask, obj, boxes, channels);
}